// AdaptiveRouter_85272280695209
// MI455X (gfx1250) — compile-verified
//
#include <hip/hip_runtime.h>
#include <hip/hip_bf16.h>

typedef float v2f __attribute__((ext_vector_type(2)));
typedef float v8f __attribute__((ext_vector_type(8)));

#define N_TOKENS  16384
#define HIDDEN    4096
#define N_EXPERTS 64
#define TOPK      8

#define GROUPS_PER_BLOCK 4           // token groups of 32 tokens each
#define K_SPLIT 2                    // waves per token group, splitting K
#define TOKENS_PER_GROUP 32          // two 16-row WMMA tiles
#define N_BLOCKS 128                 // 128 * 4 * 32 = 16384 tokens
#define K_PER_WAVE (HIDDEN / K_SPLIT)            // 2048
#define N_SCANTASKS (N_BLOCKS * GROUPS_PER_BLOCK) // 512 partial records

// ws layout (floats):
//   [0, 64)              combined bias (adaptive_bias + ema/||ema||)
//   [64, 64 + 512*128)   per-group partials: probSum[64] then counts[64]

__global__ __launch_bounds__(64)
void router_prep(const float* __restrict__ adaptive_bias,
                 const float* __restrict__ ema,
                 float* __restrict__ bias_c) {
  int e = threadIdx.x;
  float ss = 0.f;
  for (int i = 0; i < N_EXPERTS; ++i) { float v = ema[i]; ss += v * v; }
  float norm = fmaxf(sqrtf(ss), 1e-12f);
  bias_c[e] = adaptive_bias[e] + ema[e] / norm;
}

__global__ __launch_bounds__(256)
void router_main(const float* __restrict__ hidden,
                 const float* __restrict__ weight,
                 const float* __restrict__ bias_c,
                 float* __restrict__ out_w,
                 float* __restrict__ out_i,
                 float* __restrict__ partials) {
  // [kHalf][group][token][expert] : 2*4*32*64*4B = 64 KB
  __shared__ float ldsLog[K_SPLIT][GROUPS_PER_BLOCK][TOKENS_PER_GROUP][N_EXPERTS];
  __shared__ float ldsBias[N_EXPERTS];

  const int tid   = threadIdx.x;
  const int wave  = tid >> 5;          // 0..7
  const int lane  = tid & 31;
  const int group = wave & (GROUPS_PER_BLOCK - 1);   // 0..3
  const int kHalf = wave >> 2;                        // 0..1

  if (tid < N_EXPERTS) ldsBias[tid] = bias_c[tid];

  const int tokenBase = (blockIdx.x * GROUPS_PER_BLOCK + group) * TOKENS_PER_GROUP;
  const int kBase     = kHalf * K_PER_WAVE;

  // CDNA5 fp32 WMMA A-operand layout (16x4, MxK):
  //   lanes 0-15 : M = lane,     VGPR0 = K+0, VGPR1 = K+1
  //   lanes 16-31: M = lane-16,  VGPR0 = K+2, VGPR1 = K+3
  // B-operand (4x16, KxN) mirrors with N across lanes. One float2 (B64) load each.
  const int m    = lane & 15;
  const int koff = (lane >> 4) * 2;

  const float* a0 = hidden + (size_t)(tokenBase + m) * HIDDEN + kBase + koff;
  const float* a1 = a0 + (size_t)16 * HIDDEN;
  const float* bp = weight + (size_t)m * HIDDEN + kBase + koff; // expert tile t adds t*16*HIDDEN

  v8f zero = {0.f,0.f,0.f,0.f,0.f,0.f,0.f,0.f};
  v8f acc[2][4];
  #pragma unroll
  for (int h = 0; h < 2; ++h)
    #pragma unroll
    for (int t = 0; t < 4; ++t) acc[h][t] = zero;

  #pragma unroll 4
  for (int k = 0; k < K_PER_WAVE; k += 4) {
    v2f A0 = *reinterpret_cast<const v2f*>(a0 + k);
    v2f A1 = *reinterpret_cast<const v2f*>(a1 + k);
    #pragma unroll
    for (int t = 0; t < 4; ++t) {
      v2f B = *reinterpret_cast<const v2f*>(bp + (size_t)t * 16 * HIDDEN + k);
      acc[0][t] = __builtin_amdgcn_wmma_f32_16x16x4_f32(
          false, A0, false, B, (short)0, acc[0][t], false, false);
      acc[1][t] = __builtin_amdgcn_wmma_f32_16x16x4_f32(
          false, A1, false, B, (short)0, acc[1][t], false, false);
    }
  }

  // C/D layout: VGPR r holds row r (lanes 0-15) and row r+8 (lanes 16-31),
  // N = (lane&15) within each 16-expert tile.
  const int halfSel = lane >> 4;
  #pragma unroll
  for (int h = 0; h < 2; ++h)
    #pragma unroll
    for (int t = 0; t < 4; ++t)
      #pragma unroll
      for (int r = 0; r < 8; ++r) {
        int mm = h * 16 + r + 8 * halfSel;
        ldsLog[kHalf][group][mm][t * 16 + m] = acc[h][t][r];
      }

  __syncthreads();   // publish both K-half partial tiles + ldsBias

  if (kHalf != 0) return;   // K-half-1 waves are done; scan handled by K-half-0

  // ---- per-token softmax + top-8 (one lane per token, 32 tokens per scan wave) ----
  float* lo = ldsLog[0][group][lane];
  float* hi = ldsLog[1][group][lane];

  float mx = -3.4e38f;
  for (int e = 0; e < N_EXPERTS; ++e)
    mx = fmaxf(mx, lo[e] + hi[e] + ldsBias[e]);

  float tv[TOPK]; int ti[TOPK];
  #pragma unroll
  for (int j = 0; j < TOPK; ++j) { tv[j] = -1.f; ti[j] = 0; }

  float s = 0.f;
  for (int e = 0; e < N_EXPERTS; ++e) {
    float x = expf((lo[e] + hi[e] + ldsBias[e]) - mx);
    s += x;
    lo[e] = x;                          // stash exp value back in LDS
    if (x > tv[TOPK - 1]) {             // strict > keeps lower index on ties (matches top_k)
      int p = TOPK - 1;
      while (p > 0 && tv[p - 1] < x) {
        tv[p] = tv[p - 1]; ti[p] = ti[p - 1]; --p;
      }
      tv[p] = x; ti[p] = e;
    }
  }
  const float inv_s = 1.f / s;

  // outputs: weights renormalized over top-8 (== softmax-then-renorm), indices as float
  float s8 = 0.f;
  #pragma unroll
  for (int j = 0; j < TOPK; ++j) s8 += tv[j];
  const float inv8 = 1.f / s8;
  const int token = tokenBase + lane;
  #pragma unroll
  for (int j = 0; j < TOPK; ++j) {
    out_w[token * TOPK + j] = tv[j] * inv8;
    out_i[token * TOPK + j] = (float)ti[j];
  }

  // ---- deterministic per-group aux partials via fixed shuffle tree ----
  const int scanTask = blockIdx.x * GROUPS_PER_BLOCK + group;   // 0..511
  float* wpart = partials + (size_t)scanTask * 128;
  for (int e = 0; e < N_EXPERTS; ++e) {
    float pv = lo[e] * inv_s;           // this token's softmax prob for expert e
    float cv = 0.f;
    #pragma unroll
    for (int j = 0; j < TOPK; ++j) cv += (ti[j] == e) ? 1.f : 0.f;
    #pragma unroll
    for (int off = 16; off > 0; off >>= 1) {
      pv += __shfl_xor(pv, off, 32);
      cv += __shfl_xor(cv, off, 32);
    }
    if (lane == 0) {
      wpart[e]      = pv;
      wpart[64 + e] = cv;
    }
  }
}

__global__ __launch_bounds__(64)
void router_aux(const float* __restrict__ partials, float* __restrict__ aux_out) {
  __shared__ float terms[N_EXPERTS];
  const int e = threadIdx.x;
  float p = 0.f, c = 0.f;
  for (int w = 0; w < N_SCANTASKS; ++w) {          // fixed order -> deterministic
    p += partials[(size_t)w * 128 + e];
    c += partials[(size_t)w * 128 + 64 + e];
  }
  const float tokens_frac = c * (1.f / (float)(N_TOKENS * TOPK));
  const float mean_prob   = p * (1.f / (float)N_TOKENS);
  terms[e] = tokens_frac * mean_prob;
  __syncthreads();
  if (e == 0) {
    float sum = 0.f;
    for (int i = 0; i < N_EXPERTS; ++i) sum += terms[i];
    aux_out[0] = (float)N_EXPERTS * sum;
  }
}

extern "C" void kernel_launch(void* const* d_in, const int* in_sizes, int n_in,
                              void* d_out, int out_size, void* d_ws, size_t ws_size,
                              hipStream_t stream) {
  const float* hidden = (const float*)d_in[0];
  const float* weight = (const float*)d_in[1];
  const float* abias  = (const float*)d_in[2];
  const float* ema    = (const float*)d_in[3];

  float* out   = (float*)d_out;
  float* out_w = out;                                  // 16384*8
  float* out_i = out + (size_t)N_TOKENS * TOPK;        // 16384*8 (indices as float)
  float* aux   = out + (size_t)2 * N_TOKENS * TOPK;    // 1

  float* ws       = (float*)d_ws;
  float* bias_c   = ws;            // 64 floats
  float* partials = ws + 64;       // 512 * 128 floats

  router_prep<<<1, 64, 0, stream>>>(abias, ema, bias_c);
  router_main<<<N_BLOCKS, 256, 0, stream>>>(hidden, weight, bias_c, out_w, out_i, partials);
  router_aux<<<1, 64, 0, stream>>>(partials, aux);
}